// GroupGNN_15238543966901
// MI455X (gfx1250) — compile-verified
//
#include <hip/hip_runtime.h>
#include <hip/hip_bf16.h>

// Problem constants (match reference)
#define N_NODES 50000
#define N_EDGES 800000
#define HDIM    128
#define FDIM    8
#define NHEADS  4
#define EDIM    4
#define NLAYERS 2

typedef __attribute__((ext_vector_type(2))) float v2f;
typedef __attribute__((ext_vector_type(8))) float v8f;

// ---------------------------------------------------------------------------
// Generic fp32 WMMA GEMM:  C[M,Nc] = op(A[M,K] @ W[K,Nc] + bias (+resid))
// Block = 128 threads (4 waves). Block handles 16 rows x 128 cols.
// A tile (16 x K) staged in LDS via GLOBAL_LOAD_ASYNC_TO_LDS_B128 (ASYNCcnt),
// padded row stride K+4 floats: keeps 16B alignment per chunk and is LDS
// bank-conflict-free for K in {128,136,256} (4l / 12l mod 64 distinct, l<16).
// Each wave computes a 16x32 tile: two V_WMMA_F32_16X16X4_F32 accumulators
// sharing one A fragment per k-step.
// Concat-A support: k < K1 -> A (stride sA), k >= K1 -> A2 (stride sA2).
// ---------------------------------------------------------------------------
__global__ __launch_bounds__(128) void gemm_wmma_f32(
    const float* __restrict__ A, int sA, int K1,
    const float* __restrict__ A2, int sA2,
    int K, const float* __restrict__ W, int Ncols,
    const float* __restrict__ bias, const float* __restrict__ resid,
    int relu, float* __restrict__ C)
{
    extern __shared__ float As[];          // 16 * (K+4) floats
    const int ldA  = K + 4;
    const int row0 = blockIdx.x * 16;
    const int tid  = threadIdx.x;
    const int kc   = K >> 2;               // 16B chunks per row

    // async-stage the 16-row A stripe into LDS (global -> LDS, no VGPR bounce)
    for (int idx = tid; idx < 16 * kc; idx += 128) {
        int r = idx / kc, c = idx - r * kc;
        int k = c << 2;
        const float* gp = (k < K1)
            ? (A  + (size_t)(row0 + r) * sA  + k)
            : (A2 + (size_t)(row0 + r) * sA2 + (k - K1));
        unsigned ldsa = (unsigned)(size_t)(As + r * ldA + k);
        asm volatile("global_load_async_to_lds_b128 %0, %1, off"
                     :: "v"(ldsa), "v"(gp) : "memory");
    }
    asm volatile("s_wait_asynccnt 0x0" ::: "memory");
    __syncthreads();

    const int wave = tid >> 5;
    const int lane = tid & 31;
    const int half = lane >> 4;            // 0: K{0,1}/rows 0..7 ; 1: K{2,3}/rows 8..15
    const int lr   = lane & 15;
    const int c0   = blockIdx.y * 128 + wave * 32;
    const int col0 = c0 + lr;
    const int col1 = c0 + 16 + lr;

    const float* wp0 = W + col0;
    const float* wp1 = W + col1;
    __builtin_prefetch(wp0, 0, 1);         // global_prefetch_b8 on the weight panel

    v8f acc0 = {0.f, 0.f, 0.f, 0.f, 0.f, 0.f, 0.f, 0.f};
    v8f acc1 = {0.f, 0.f, 0.f, 0.f, 0.f, 0.f, 0.f, 0.f};
    for (int k = 0; k < K; k += 4) {
        v2f a, b0, b1;
        // A 16x4 fragment: lanes 0-15 hold K=k,k+1 ; lanes 16-31 hold K=k+2,k+3
        a.x = As[lr * ldA + k + 2 * half];
        a.y = As[lr * ldA + k + 2 * half + 1];
        // B 4x16 fragments (two col tiles reuse the same A fragment)
        b0.x = wp0[(size_t)(k + 2 * half)     * Ncols];
        b0.y = wp0[(size_t)(k + 2 * half + 1) * Ncols];
        b1.x = wp1[(size_t)(k + 2 * half)     * Ncols];
        b1.y = wp1[(size_t)(k + 2 * half + 1) * Ncols];
        acc0 = __builtin_amdgcn_wmma_f32_16x16x4_f32(
                   false, a, false, b0, (short)0, acc0, false, false);
        acc1 = __builtin_amdgcn_wmma_f32_16x16x4_f32(
                   false, a, false, b1, (short)0, acc1, false, false);
    }

    const float bc0 = bias ? bias[col0] : 0.f;
    const float bc1 = bias ? bias[col1] : 0.f;
#pragma unroll
    for (int j = 0; j < 8; ++j) {          // VGPR j: lanes<16 -> M=j, lanes>=16 -> M=j+8
        int row = row0 + j + 8 * half;
        size_t o0 = (size_t)row * Ncols + col0;
        size_t o1 = (size_t)row * Ncols + col1;
        float v0 = acc0[j] + bc0;
        float v1 = acc1[j] + bc1;
        if (relu) { v0 = fmaxf(v0, 0.f); v1 = fmaxf(v1, 0.f); }
        if (resid) { v0 += resid[o0]; v1 += resid[o1]; }
        C[o0] = v0;
        C[o1] = v1;
    }
}

// ---------------------------------------------------------------------------
// Per-layer scratch init: aggr = 0, m = -inf, denom = 0
// ---------------------------------------------------------------------------
__global__ __launch_bounds__(256) void init_kernel(
    float* __restrict__ aggr, float* __restrict__ m, float* __restrict__ denom)
{
    size_t idx = (size_t)blockIdx.x * 256 + threadIdx.x;
    if (idx < (size_t)N_NODES * HDIM) aggr[idx] = 0.f;
    if (idx < (size_t)N_NODES * NHEADS) { m[idx] = -INFINITY; denom[idx] = 0.f; }
}

// ---------------------------------------------------------------------------
// Attention scores: one wave per (edge, head). Lane = one of 32 head dims.
// attn[e,h] = dot(q[dst], k[src]) / eff_scale + edge_attr @ Wep + bep
// Also segment-max via atomicMax(float).
// ---------------------------------------------------------------------------
__global__ __launch_bounds__(256) void attn_kernel(
    const float* __restrict__ q, const float* __restrict__ k,
    const float* __restrict__ ea, const float* __restrict__ Wep,
    const float* __restrict__ bep, const float* __restrict__ scale, int layer,
    const int* __restrict__ src, const int* __restrict__ dst,
    float* __restrict__ attn, float* __restrict__ mmax)
{
    int gw = blockIdx.x * 8 + (threadIdx.x >> 5);   // global wave id
    int e  = gw >> 2;
    int hh = gw & 3;
    if (e >= N_EDGES) return;
    int lane = threadIdx.x & 31;
    int s = src[e], d = dst[e];

    float v = q[(size_t)d * HDIM + hh * 32 + lane] *
              k[(size_t)s * HDIM + hh * 32 + lane];
    for (int off = 16; off > 0; off >>= 1) v += __shfl_down(v, off, 32);

    if (lane == 0) {
        float eff = fmaxf(fabsf(scale[layer]), 0.01f) * 5.656854249492380f; // sqrt(32)
        float a = v / eff;
        const float* wep = Wep + layer * EDIM * NHEADS;
#pragma unroll
        for (int dd = 0; dd < EDIM; ++dd)
            a += ea[(size_t)e * EDIM + dd] * wep[dd * NHEADS + hh];
        a += bep[layer * NHEADS + hh];
        attn[(size_t)e * NHEADS + hh] = a;
        atomicMax(&mmax[(size_t)d * NHEADS + hh], a);
    }
}

// ---------------------------------------------------------------------------
// ex = exp(attn - m[dst]); denom[dst] += ex   (one thread per (edge, head))
// ---------------------------------------------------------------------------
__global__ __launch_bounds__(256) void exp_kernel(
    float* __restrict__ attn, const float* __restrict__ m,
    float* __restrict__ denom, const int* __restrict__ dst)
{
    size_t idx = (size_t)blockIdx.x * 256 + threadIdx.x;
    if (idx >= (size_t)N_EDGES * NHEADS) return;
    int e = (int)(idx >> 2), hh = (int)(idx & 3);
    int d = dst[e];
    float ex = expf(attn[idx] - m[(size_t)d * NHEADS + hh]);
    attn[idx] = ex;
    atomicAdd(&denom[(size_t)d * NHEADS + hh], ex);
}

// ---------------------------------------------------------------------------
// alpha = ex / (denom[dst]+eps); write alphas; aggr[dst] += alpha * v[src]
// One wave per edge; lane covers 32 features per head chunk (4 chunks = H).
// ---------------------------------------------------------------------------
__global__ __launch_bounds__(256) void msg_kernel(
    const float* __restrict__ ex, const float* __restrict__ denom,
    const float* __restrict__ vmat, const int* __restrict__ src,
    const int* __restrict__ dst, float* __restrict__ alpha_out,
    float* __restrict__ aggr)
{
    int e = blockIdx.x * 8 + (threadIdx.x >> 5);
    if (e >= N_EDGES) return;
    int lane = threadIdx.x & 31;
    int s = src[e], d = dst[e];

    if (lane < NHEADS)
        alpha_out[(size_t)e * NHEADS + lane] =
            ex[(size_t)e * NHEADS + lane] /
            (denom[(size_t)d * NHEADS + lane] + 1e-16f);

#pragma unroll
    for (int c = 0; c < NHEADS; ++c) {
        float al = ex[(size_t)e * NHEADS + c] /
                   (denom[(size_t)d * NHEADS + c] + 1e-16f);
        float val = al * vmat[(size_t)s * HDIM + c * 32 + lane];
        atomicAdd(&aggr[(size_t)d * HDIM + c * 32 + lane], val);
    }
}

// ---------------------------------------------------------------------------
// LayerNorm over H=128: one wave per row, 4 elems/lane, shfl_xor reduction.
// Matches jnp: population variance, eps=1e-5.
// ---------------------------------------------------------------------------
__global__ __launch_bounds__(256) void ln_kernel(
    const float* __restrict__ in, const float* __restrict__ g,
    const float* __restrict__ b, float* __restrict__ out)
{
    int row = blockIdx.x * 8 + (threadIdx.x >> 5);
    if (row >= N_NODES) return;
    int lane = threadIdx.x & 31;
    const float* ip = in + (size_t)row * HDIM;
    float x0 = ip[lane], x1 = ip[lane + 32], x2 = ip[lane + 64], x3 = ip[lane + 96];
    float s  = x0 + x1 + x2 + x3;
    float ss = x0 * x0 + x1 * x1 + x2 * x2 + x3 * x3;
    for (int off = 16; off > 0; off >>= 1) {
        s  += __shfl_xor(s,  off, 32);
        ss += __shfl_xor(ss, off, 32);
    }
    float mu  = s * (1.f / 128.f);
    float var = ss * (1.f / 128.f) - mu * mu;
    float rs  = rsqrtf(var + 1e-5f);
    float* op = out + (size_t)row * HDIM;
    op[lane]      = (x0 - mu) * rs * g[lane]      + b[lane];
    op[lane + 32] = (x1 - mu) * rs * g[lane + 32] + b[lane + 32];
    op[lane + 64] = (x2 - mu) * rs * g[lane + 64] + b[lane + 64];
    op[lane + 96] = (x3 - mu) * rs * g[lane + 96] + b[lane + 96];
}

// ---------------------------------------------------------------------------
extern "C" void kernel_launch(void* const* d_in, const int* in_sizes, int n_in,
                              void* d_out, int out_size, void* d_ws, size_t ws_size,
                              hipStream_t stream)
{
    const float* team  = (const float*)d_in[0];
    const float* x     = (const float*)d_in[1];
    const float* ea    = (const float*)d_in[2];
    const float* Wi    = (const float*)d_in[3];
    const float* bi    = (const float*)d_in[4];
    const float* Wq    = (const float*)d_in[5];
    const float* Wk    = (const float*)d_in[6];
    const float* Wv    = (const float*)d_in[7];
    const float* Wep   = (const float*)d_in[8];
    const float* bep   = (const float*)d_in[9];
    const float* scale = (const float*)d_in[10];
    const float* Wo    = (const float*)d_in[11];
    const float* bo    = (const float*)d_in[12];
    const float* ln_g  = (const float*)d_in[13];
    const float* ln_b  = (const float*)d_in[14];
    const float* W1    = (const float*)d_in[15];
    const float* b1    = (const float*)d_in[16];
    const float* W2    = (const float*)d_in[17];
    const float* b2    = (const float*)d_in[18];
    const float* fn_g  = (const float*)d_in[19];
    const float* fn_b  = (const float*)d_in[20];
    const int*   eidx  = (const int*)d_in[21];
    const int*   src   = eidx;
    const int*   dst   = eidx + N_EDGES;

    // workspace carve-out (floats)
    float* ws = (float*)d_ws;
    size_t o = 0;
    float* h    = ws + o; o += (size_t)N_NODES * HDIM;
    float* qb   = ws + o; o += (size_t)N_NODES * HDIM;   // qb..kb contiguous ->
    float* kb   = ws + o; o += (size_t)N_NODES * HDIM;   //   reused as FFN hidden [N,256]
    float* vb   = ws + o; o += (size_t)N_NODES * HDIM;
    float* aggr = ws + o; o += (size_t)N_NODES * HDIM;
    float* tmp  = ws + o; o += (size_t)N_NODES * HDIM;
    float* exb  = ws + o; o += (size_t)N_EDGES * NHEADS;
    float* mb   = ws + o; o += (size_t)N_NODES * NHEADS;
    float* db   = ws + o; o += (size_t)N_NODES * NHEADS;
    float* ffn  = qb;                                     // [N, 2H]

    float* out_h     = (float*)d_out;
    float* out_alpha = out_h + (size_t)N_NODES * HDIM;    // [L, E, NH]

    const dim3 blk(256);
    const dim3 gblk(128);
    auto gemm = [&](const float* A, int sA, int K1, const float* A2, int sA2, int K,
                    const float* W, int Nc, const float* bias, const float* resid,
                    int relu, float* C) {
        dim3 grid(N_NODES / 16, Nc / 128);
        size_t shmem = (size_t)16 * (K + 4) * sizeof(float);
        gemm_wmma_f32<<<grid, gblk, shmem, stream>>>(A, sA, K1, A2, sA2, K, W, Nc,
                                                     bias, resid, relu, C);
    };

    // h = [team | x] @ Wi + bi
    gemm(team, HDIM, HDIM, x, FDIM, HDIM + FDIM, Wi, HDIM, bi, nullptr, 0, h);

    for (int i = 0; i < NLAYERS; ++i) {
        const size_t HH = (size_t)HDIM * HDIM;
        // node-level Q/K/V projections
        gemm(h, HDIM, HDIM, h, HDIM, HDIM, Wq + i * HH, HDIM, nullptr, nullptr, 0, qb);
        gemm(h, HDIM, HDIM, h, HDIM, HDIM, Wk + i * HH, HDIM, nullptr, nullptr, 0, kb);
        gemm(h, HDIM, HDIM, h, HDIM, HDIM, Wv + i * HH, HDIM, nullptr, nullptr, 0, vb);

        // edge phase
        init_kernel<<<(N_NODES * HDIM) / 256, blk, 0, stream>>>(aggr, mb, db);
        attn_kernel<<<(N_EDGES * NHEADS) / 8, blk, 0, stream>>>(
            qb, kb, ea, Wep, bep, scale, i, src, dst, exb, mb);
        exp_kernel<<<((size_t)N_EDGES * NHEADS + 255) / 256, blk, 0, stream>>>(
            exb, mb, db, dst);
        msg_kernel<<<N_EDGES / 8, blk, 0, stream>>>(
            exb, db, vb, src, dst, out_alpha + (size_t)i * N_EDGES * NHEADS, aggr);

        // h = LN(h + aggr @ Wo + bo)
        gemm(aggr, HDIM, HDIM, aggr, HDIM, HDIM, Wo + i * HH, HDIM, bo + i * HDIM,
             h, 0, tmp);
        ln_kernel<<<N_NODES / 8, blk, 0, stream>>>(tmp, ln_g + i * HDIM,
                                                   ln_b + i * HDIM, h);

        // h = h + relu(h @ W1 + b1) @ W2 + b2
        gemm(h, HDIM, HDIM, h, HDIM, HDIM, W1 + (size_t)i * HDIM * 2 * HDIM,
             2 * HDIM, b1 + i * 2 * HDIM, nullptr, 1, ffn);
        gemm(ffn, 2 * HDIM, 2 * HDIM, ffn, 2 * HDIM, 2 * HDIM,
             W2 + (size_t)i * 2 * HDIM * HDIM, HDIM, b2 + i * HDIM, h, 0, h);
    }

    // final LN -> output
    ln_kernel<<<N_NODES / 8, blk, 0, stream>>>(h, fn_g, fn_b, out_h);
}